// Unsuper_Cross_Entropy_40973988004534
// MI455X (gfx1250) — compile-verified
//
#include <hip/hip_runtime.h>
#include <hip/hip_bf16.h>
#include <math.h>

typedef __attribute__((ext_vector_type(2))) float v2f;
typedef __attribute__((ext_vector_type(8))) float v8f;

#define D_FEAT          512
#define TILE            16
#define WAVES_PER_BLOCK 4
#define NUM_BLOCKS      256
#define THREADS         (WAVES_PER_BLOCK * 32)

// Each wave computes 16 edge dot-products via V_WMMA_F32_16X16X4_F32:
// A = 16 gathered src rows (chunked K=4), B = 16 gathered dst rows (transposed),
// diag(A x B^T) = per-edge scores. Then stable BCE-with-logits, partial sums out.
__global__ __launch_bounds__(THREADS)
void edge_bce_wmma_kernel(const float* __restrict__ feat,
                          const int* __restrict__ pos_src,
                          const int* __restrict__ pos_dst,
                          const int* __restrict__ neg_src,
                          const int* __restrict__ neg_dst,
                          int Epos, int Etot,
                          float* __restrict__ partials)
{
    __shared__ float wave_part[WAVES_PER_BLOCK];

    const int lane  = threadIdx.x & 31;
    const int wave  = threadIdx.x >> 5;
    const int gwave = blockIdx.x * WAVES_PER_BLOCK + wave;
    const int nwave = gridDim.x * WAVES_PER_BLOCK;
    const int ntile = (Etot + TILE - 1) / TILE;

    const int l16 = lane & 15;      // which edge's row this lane fetches
    const int h   = lane >> 4;      // K half: 0 -> K{0,1}, 1 -> K{2,3}
    // Diagonal ownership of the 16x16 f32 accumulator (wave32 layout):
    // lanes 0..7 own edges 0..7 (acc[lane]); lanes 24..31 own edges 8..15 (acc[lane-24]).
    const int sel  = (lane < 8) ? lane : ((lane >= 24) ? (lane - 24) : -1);
    const int eown = (lane < 8) ? lane : (lane - 16);

    float lsum = 0.0f;

    for (int t = gwave; t < ntile; t += nwave) {
        // Gather node indices for this tile's 16 edges (lanes l and l+16 duplicate).
        int ge_row = t * TILE + l16;
        int sidx = 0, didx = 0;
        if (ge_row < Etot) {
            if (ge_row < Epos) {
                sidx = pos_src[ge_row];
                didx = pos_dst[ge_row];
            } else {
                int g = ge_row - Epos;
                sidx = neg_src[g];
                didx = neg_dst[g];
            }
        }
        const v2f* __restrict__ urow =
            (const v2f*)(feat + (long)sidx * D_FEAT + 2 * h);
        const v2f* __restrict__ vrow =
            (const v2f*)(feat + (long)didx * D_FEAT + 2 * h);

        v8f acc = {};
        #pragma unroll 8
        for (int c = 0; c < D_FEAT / 4; ++c) {
            v2f a = urow[2 * c];   // features 4c+2h, 4c+2h+1 of src row
            v2f b = vrow[2 * c];   // features 4c+2h, 4c+2h+1 of dst row
            acc = __builtin_amdgcn_wmma_f32_16x16x4_f32(
                false, a, false, b, (short)0, acc, false, false);
        }

        // Extract owned diagonal element (score for edge `eown`).
        float s = 0.0f;
        #pragma unroll
        for (int j = 0; j < 8; ++j)
            if (sel == j) s = acc[j];

        int ge = t * TILE + eown;
        if (sel >= 0 && ge < Etot) {
            float label = (ge < Epos) ? 1.0f : 0.0f;
            // numerically-stable BCE with logits
            lsum += fmaxf(s, 0.0f) - s * label + log1pf(expf(-fabsf(s)));
        }
    }

    // Deterministic wave reduction (wave32).
    #pragma unroll
    for (int off = 16; off > 0; off >>= 1)
        lsum += __shfl_down(lsum, off, 32);

    if (lane == 0) wave_part[wave] = lsum;
    __syncthreads();
    if (threadIdx.x == 0) {
        float b = 0.0f;
        #pragma unroll
        for (int w = 0; w < WAVES_PER_BLOCK; ++w) b += wave_part[w];
        partials[blockIdx.x] = b;
    }
}

__global__ __launch_bounds__(256)
void final_reduce_kernel(const float* __restrict__ partials, int n,
                         float* __restrict__ out, float inv_count)
{
    __shared__ float sdata[256];
    int t = threadIdx.x;
    float v = 0.0f;
    for (int i = t; i < n; i += 256) v += partials[i];
    sdata[t] = v;
    __syncthreads();
    #pragma unroll
    for (int s = 128; s > 0; s >>= 1) {
        if (t < s) sdata[t] += sdata[t + s];
        __syncthreads();
    }
    if (t == 0) out[0] = sdata[0] * inv_count;
}

extern "C" void kernel_launch(void* const* d_in, const int* in_sizes, int n_in,
                              void* d_out, int out_size, void* d_ws, size_t ws_size,
                              hipStream_t stream) {
    const float* feat   = (const float*)d_in[0];
    const int* pos_src  = (const int*)d_in[1];
    const int* pos_dst  = (const int*)d_in[2];
    const int* neg_src  = (const int*)d_in[3];
    const int* neg_dst  = (const int*)d_in[4];

    const int Epos = in_sizes[1];
    const int Eneg = in_sizes[3];
    const int Etot = Epos + Eneg;

    float* partials = (float*)d_ws;   // NUM_BLOCKS floats of scratch

    edge_bce_wmma_kernel<<<NUM_BLOCKS, THREADS, 0, stream>>>(
        feat, pos_src, pos_dst, neg_src, neg_dst, Epos, Etot, partials);

    final_reduce_kernel<<<1, 256, 0, stream>>>(
        partials, NUM_BLOCKS, (float*)d_out, 1.0f / (float)Etot);
}